// GaussSoftmaxV3_17583596109902
// MI455X (gfx1250) — compile-verified
//
#include <hip/hip_runtime.h>
#include <math.h>

// Problem constants (from reference)
#define V 32000
#define T 128
#define DD 64
#define NN 512

#define BND     1.41421356237309515f   // sqrt(2)
#define LSLO   -2.30258509299404590f   // log(0.1)
#define LSHI    2.30258509299404590f   // log(10)
#define LOG2PI  1.83787706640934534f   // log(2*pi)

typedef __attribute__((ext_vector_type(2))) float v2f;
typedef __attribute__((ext_vector_type(8))) float v8f;

__device__ __forceinline__ float clampf(float x, float lo, float hi) {
    return fminf(fmaxf(x, lo), hi);
}

// ---------------------------------------------------------------------------
// Kernel 0: per-(t,d) tables: clamped mu_c, sigma_c=exp(clamp ls_c), 1/sigma_c,
// and per-t sum of clamped log_sigma_c. grid=T blocks of D threads.
// ---------------------------------------------------------------------------
__global__ void k_prep(const float* __restrict__ mu_c, const float* __restrict__ ls_c,
                       float* __restrict__ tmu, float* __restrict__ tsc,
                       float* __restrict__ tis, float* __restrict__ slc) {
    __shared__ float red[DD];
    int t = blockIdx.x, d = threadIdx.x;
    int g = t * DD + d;
    float m = clampf(mu_c[g], -BND, BND);
    float l = clampf(ls_c[g], LSLO, LSHI);
    float s = __expf(l);
    tmu[g] = m;
    tsc[g] = s;
    tis[g] = 1.0f / s;
    red[d] = l;
    __syncthreads();
    for (int k = DD / 2; k > 0; k >>= 1) {
        if (d < k) red[d] += red[d + k];
        __syncthreads();
    }
    if (d == 0) slc[t] = red[0];
}

// ---------------------------------------------------------------------------
// Kernel 1: fused energy + KL. One thread per vocab row v; mu_v/sigma_v kept
// in VGPRs (fully unrolled d-loop); per-t tables broadcast from LDS (uniform
// address across the wave -> conflict-free). LDS staged in two 64-t chunks
// (3 * 16KB = 48KB) to stay under static-LDS limits.
// ---------------------------------------------------------------------------
__global__ void k_energy(const float* __restrict__ mu, const float* __restrict__ lsg,
                         const float* __restrict__ tmu, const float* __restrict__ tsc,
                         const float* __restrict__ tis, const float* __restrict__ slc,
                         float* __restrict__ e_ws, float* __restrict__ kl_out) {
    __shared__ float s_mu[64 * DD];
    __shared__ float s_sc[64 * DD];
    __shared__ float s_is[64 * DD];
    __shared__ float s_sl[T];

    int tid = threadIdx.x;            // blockDim.x == 128
    s_sl[tid] = slc[tid];

    int v = blockIdx.x * 128 + tid;
    float muv[DD], sv[DD];
    float sum_lsv = 0.0f;
#pragma unroll
    for (int d = 0; d < DD; ++d) {
        float m = clampf(mu[v * DD + d], -BND, BND);
        float l = clampf(lsg[v * DD + d], LSLO, LSHI);
        muv[d] = m;
        sv[d]  = __expf(l);
        sum_lsv += l;
    }

    for (int tc = 0; tc < 2; ++tc) {
        __syncthreads();              // protect prior chunk's readers
        for (int i = tid; i < 64 * DD; i += 128) {
            int g = tc * 64 * DD + i;
            s_mu[i] = tmu[g];
            s_sc[i] = tsc[g];
            s_is[i] = tis[g];
        }
        __syncthreads();
        for (int tt = 0; tt < 64; ++tt) {
            int t = tc * 64 + tt;
            float det = 0.0f, dme = 0.0f, tr = 0.0f, dmk = 0.0f;
#pragma unroll
            for (int d = 0; d < DD; ++d) {
                float st  = s_sc[tt * DD + d];
                float ist = s_is[tt * DD + d];
                float mtd = s_mu[tt * DD + d];
                float s   = sv[d] + st;
                float dmu = muv[d] - mtd;
                float dm2 = dmu * dmu;
                float rs  = __builtin_amdgcn_rcpf(s);
                det += __logf(s);
                dme += dm2 * rs;
                tr  += sv[d] * ist;
                dmk += dm2 * ist;
            }
            float energy = -0.5f * (det + dme + (float)DD * LOG2PI);
            float kl = 0.5f * (s_sl[t] - sum_lsv - (float)DD + tr + dmk);
            e_ws[(size_t)v * T + t]  = energy;
            kl_out[(size_t)v * T + t] = kl;
        }
    }
}

// ---------------------------------------------------------------------------
// Kernel 2: column softmax over V. One block per column t. Pass 1: max.
// Pass 2: rewrite workspace in place with w = exp(e - m), accumulate Z,
// emit invZ[t]. grid=T blocks of 256 threads.
// ---------------------------------------------------------------------------
__global__ void k_softmax_col(float* __restrict__ e_ws, float* __restrict__ invZ) {
    __shared__ float red[256];
    int t = blockIdx.x, tid = threadIdx.x;
    float m = -3.0e38f;
    for (int v = tid; v < V; v += 256) m = fmaxf(m, e_ws[(size_t)v * T + t]);
    red[tid] = m;
    __syncthreads();
    for (int k = 128; k > 0; k >>= 1) {
        if (tid < k) red[tid] = fmaxf(red[tid], red[tid + k]);
        __syncthreads();
    }
    m = red[0];
    __syncthreads();
    float sum = 0.0f;
    for (int v = tid; v < V; v += 256) {
        float wv = __expf(e_ws[(size_t)v * T + t] - m);
        e_ws[(size_t)v * T + t] = wv;
        sum += wv;
    }
    red[tid] = sum;
    __syncthreads();
    for (int k = 128; k > 0; k >>= 1) {
        if (tid < k) red[tid] += red[tid + k];
        __syncthreads();
    }
    if (tid == 0) invZ[t] = 1.0f / red[0];
}

// ---------------------------------------------------------------------------
// Kernel 2b: xs[t][n] = x[t][n] * invZ[t]   (folds softmax normalization
// into the GEMM B operand). 65536 elements.
// ---------------------------------------------------------------------------
__global__ void k_scale_x(const float* __restrict__ x, const float* __restrict__ invZ,
                          float* __restrict__ xs) {
    int i = blockIdx.x * 256 + threadIdx.x;
    xs[i] = x[i] * invZ[i >> 9];   // NN = 512
}

// ---------------------------------------------------------------------------
// Kernel 3: out = w @ xs via v_wmma_f32_16x16x4_f32. One 16x16 output tile
// per wave32; K = 128 in steps of 4. Operand placement per CDNA5 ISA 7.12.2:
//   A 16x4 : lane<16 -> M=lane, K={0,1}; lane>=16 -> M=lane-16, K={2,3}
//   B 4x16 : lane<16 -> N=lane, K={0,1}; lane>=16 -> N=lane-16, K={2,3}
//   C 16x16: VGPR r -> M = r + 8*(lane>=16), N = lane%16
// Block = 256 threads (8 waves) covers a 16(M) x 128(N) block tile.
// ---------------------------------------------------------------------------
__global__ void k_gemm(const float* __restrict__ wmat, const float* __restrict__ xs,
                       float* __restrict__ out) {
    int lane = threadIdx.x & 31;
    int wv   = threadIdx.x >> 5;       // wave id 0..7
    int half = lane >> 4;              // 0/1 -> K pair select, M half select
    int lm   = lane & 15;
    int v0   = blockIdx.x * 16;
    int n0   = blockIdx.y * 128 + wv * 16;

    const float* arow = wmat + (size_t)(v0 + lm) * T + half * 2;
    const float* bcol = xs + (size_t)(half * 2) * NN + n0 + lm;

    v8f acc = {};
#pragma unroll 4
    for (int k0 = 0; k0 < T; k0 += 4) {
        __builtin_prefetch(arow + k0 + 32, 0, 1);         // global_prefetch_b8
        v2f a = *(const v2f*)(arow + k0);                 // K=k0+ks, k0+ks+1
        v2f b;
        b.x = bcol[(size_t)k0 * NN];                      // K=k0+ks
        b.y = bcol[(size_t)(k0 + 1) * NN];                // K=k0+ks+1
        acc = __builtin_amdgcn_wmma_f32_16x16x4_f32(
            false, a, false, b, (short)0, acc, false, false);
    }

    float* orow = out + n0 + lm;
#pragma unroll
    for (int r = 0; r < 8; ++r) {
        orow[(size_t)(v0 + r + half * 8) * NN] = acc[r];
    }
}

// ---------------------------------------------------------------------------
// Host launch
// ---------------------------------------------------------------------------
extern "C" void kernel_launch(void* const* d_in, const int* in_sizes, int n_in,
                              void* d_out, int out_size, void* d_ws, size_t ws_size,
                              hipStream_t stream) {
    const float* x    = (const float*)d_in[0];   // (T, N)
    const float* mu   = (const float*)d_in[1];   // (V, D)
    const float* lsg  = (const float*)d_in[2];   // (V, D)
    const float* mu_c = (const float*)d_in[3];   // (T, D)
    const float* ls_c = (const float*)d_in[4];   // (T, D)
    // d_in[5] = t (scalar); forward output is identical for any t -> ignored.

    float* out = (float*)d_out;                  // (V, N)
    float* kl  = out + (size_t)V * NN;           // (V, T)

    float* ws   = (float*)d_ws;
    float* e    = ws;                            // V*T   energy -> then w
    float* tmu  = e + (size_t)V * T;             // T*D
    float* tsc  = tmu + T * DD;                  // T*D
    float* tis  = tsc + T * DD;                  // T*D
    float* slc  = tis + T * DD;                  // T
    float* invZ = slc + T;                       // T
    float* xs   = invZ + T;                      // T*N

    k_prep<<<T, DD, 0, stream>>>(mu_c, ls_c, tmu, tsc, tis, slc);
    k_energy<<<V / 128, 128, 0, stream>>>(mu, lsg, tmu, tsc, tis, slc, e, kl);
    k_softmax_col<<<T, 256, 0, stream>>>(e, invZ);
    k_scale_x<<<(T * NN) / 256, 256, 0, stream>>>(x, invZ, xs);
    k_gemm<<<dim3(V / 16, NN / 128), 256, 0, stream>>>(e, xs, out);
}